// ParametricSurvivalModel_51737176047793
// MI455X (gfx1250) — compile-verified
//
#include <hip/hip_runtime.h>
#include <math.h>

// FM survival model, MI455X (gfx1250, wave32).
// Memory-bound gather kernel: each wave owns one batch row at a time; each
// lane owns two of the K=64 factor columns so every gathered weight row is a
// single fully-coalesced 256B global_load_b64 across the wave.
// The FM s.s term is computed with chained V_WMMA_F32_16X16X4_F32 (diag of
// S*S^T), which is semantically exact.

typedef __attribute__((ext_vector_type(2))) float v2f;
typedef __attribute__((ext_vector_type(8))) float v8f;

#define ROWS_PER_BLOCK 16
#define SSTRIDE 66  // 64 + 2 floats pad to spread LDS banks

__global__ __launch_bounds__(256)
void fm_survival_kernel(const int* __restrict__ featidx,
                        const float* __restrict__ featval,
                        const float* __restrict__ hist,
                        const float* __restrict__ wl,
                        const float* __restrict__ wf,
                        const float* __restrict__ intercept_p,
                        const float* __restrict__ shape_p,
                        float* __restrict__ out,
                        int B, int F) {
  __shared__ float sS[ROWS_PER_BLOCK * SSTRIDE];
  __shared__ float rssq[ROWS_PER_BLOCK];
  __shared__ float rwl[ROWS_PER_BLOCK];

  const int lane = threadIdx.x & 31;
  const int wave = threadIdx.x >> 5;
  const int blockRow = blockIdx.x * ROWS_PER_BLOCK;

  // ---- main gather/accumulate: 2 rows per wave ----
  for (int r = 0; r < 2; ++r) {
    const int rowInBlock = wave * 2 + r;
    const int b = blockRow + rowInBlock;
    float accx = 0.f, accy = 0.f, ssq = 0.f, wlsum = 0.f;
    if (b < B) {
      const long long rowBase = (long long)b * F;
      for (int base = 0; base < F; base += 32) {
        const int f = base + lane;
        int idxv = 0;
        float valv = 0.f;
        if (f < F) {
          idxv = featidx[rowBase + f];
          valv = featval[rowBase + f];
        }
        // linear term gather (valv==0 for padding -> exact no-op)
        wlsum += wl[idxv] * valv;
        // factorized gather: broadcast (idx,val) lane-by-lane, coalesced
        // 256B row load per iteration (float2 per lane over 32 lanes).
#pragma unroll 8
        for (int j = 0; j < 32; ++j) {
          const int bidx = __shfl(idxv, j, 32);
          const float bval = __shfl(valv, j, 32);
          const float2 w =
              *(const float2*)(wf + ((long long)bidx << 6) + (lane << 1));
          const float vx = w.x * bval;
          const float vy = w.y * bval;
          accx += vx;
          accy += vy;
          ssq += vx * vx + vy * vy;
        }
      }
    }
    // wave reductions for scalar terms
#pragma unroll
    for (int off = 16; off > 0; off >>= 1) {
      ssq += __shfl_down(ssq, off, 32);
      wlsum += __shfl_down(wlsum, off, 32);
    }
    // stage s-vector (row of S) + scalars in LDS
    *(float2*)&sS[rowInBlock * SSTRIDE + (lane << 1)] = make_float2(accx, accy);
    if (lane == 0) {
      rssq[rowInBlock] = ssq;
      rwl[rowInBlock] = wlsum;
    }
  }
  __syncthreads();

  // ---- epilogue on wave 0: diag(S*S^T) via 16 chained f32 WMMAs ----
  if (wave == 0) {
    const int m = lane & 15;   // A: row, B: col (register images identical)
    const int hi = lane >> 4;  // lanes 16-31 hold K=2,3 of each 4-chunk
    v8f c = {};
#pragma unroll
    for (int kc = 0; kc < 16; ++kc) {
      const float2 w = *(const float2*)&sS[m * SSTRIDE + kc * 4 + (hi << 1)];
      v2f ab;
      ab.x = w.x;
      ab.y = w.y;
      // D = A(16x4) * B(4x16) + C ; A == B register image => D = S*S^T chunk
      c = __builtin_amdgcn_wmma_f32_16x16x4_f32(
          false, ab, false, ab, (short)0, c, false, false);
    }
    float carr[8];
#pragma unroll
    for (int i = 0; i < 8; ++i) carr[i] = c[i];
    // C/D layout: row m<8 -> lane m, vgpr m ; row m>=8 -> lane m+16, vgpr m-8
    float diag = 0.f;
#pragma unroll
    for (int i = 0; i < 8; ++i) {
      if (lane == i || lane == 24 + i) diag = carr[i];
    }
    const int dm = (lane < 8) ? lane : ((lane >= 24) ? (lane - 16) : -1);
    if (dm >= 0) {
      const int row = blockRow + dm;
      if (row < B) {
        const float x = rwl[dm] + intercept_p[0] + 0.5f * (diag - rssq[dm]);
        const float sc = (x > 20.f) ? x : log1pf(expf(x));  // softplus
        const float t = hist[row];
        const float p = 1.f / (1.f + powf(t / sc, -shape_p[0]));
        out[row] = p;
        out[B + row] = (p >= 0.5f) ? 1.f : 0.f;
      }
    }
  }
}

extern "C" void kernel_launch(void* const* d_in, const int* in_sizes, int n_in,
                              void* d_out, int out_size, void* d_ws,
                              size_t ws_size, hipStream_t stream) {
  (void)d_ws;
  (void)ws_size;
  (void)out_size;
  (void)n_in;
  const int* featidx = (const int*)d_in[0];
  const float* featval = (const float*)d_in[1];
  const float* hist = (const float*)d_in[2];
  const float* wl = (const float*)d_in[3];
  const float* wf = (const float*)d_in[4];
  const float* intercept_p = (const float*)d_in[5];
  const float* shape_p = (const float*)d_in[6];
  float* out = (float*)d_out;

  const int B = in_sizes[2];            // hist_reserve_prices is [B]
  const int F = in_sizes[0] / B;        // featidx is [B, F]

  dim3 grid((B + ROWS_PER_BLOCK - 1) / ROWS_PER_BLOCK);
  dim3 block(256);
  fm_survival_kernel<<<grid, block, 0, stream>>>(
      featidx, featval, hist, wl, wf, intercept_p, shape_p, out, B, F);
}